// CPDist_73796128080636
// MI455X (gfx1250) — compile-verified
//
#include <hip/hip_runtime.h>
#include <math.h>

typedef float v2f   __attribute__((ext_vector_type(2)));
typedef float v8f   __attribute__((ext_vector_type(8)));
typedef float f32x4 __attribute__((ext_vector_type(4)));

#define B_ 8
#define T_ 128
#define D_ 1024
#define V_ 4096
#define R_ 16
#define H_ 2

// LDS row stride (floats): 1032 % 64 == 8 -> 8 rows x 8-bank groups cover all
// 64 banks exactly once for a 32-lane b128 access (lanes m and m+8 broadcast).
#define AS_STRIDE 1032

// Workspace layout (floats): [0..255] = S[h][b][r], [256..511] = E[h][b][r]

__global__ void cp_zero_ws(float* ws) {
    int t = blockIdx.x * blockDim.x + threadIdx.x;
    if (t < 512) ws[t] = 0.0f;
}

__global__ __launch_bounds__(256)
void cp_main(const float* __restrict__ lhs,        // (B, T, D)
             const float* __restrict__ w,          // (V*R*H, D)
             const float* __restrict__ bias,       // (V*R*H)
             const long long* __restrict__ points, // (B, H)
             float* __restrict__ ws)
{
    __shared__ float As[8 * AS_STRIDE];   // h_last staged, padded rows
    __shared__ float Sred[128];           // per-block S accumulator [b][r]

    const int tid  = threadIdx.x;
    const int lane = tid & 31;
    const int wave = tid >> 5;

    // Stage h_last = lhs[:, T-1, :] (8 rows x 1024 f32 = 32KB) into LDS, coalesced.
    for (int idx = tid; idx < 8 * 1024; idx += 256) {
        int m = idx >> 10;
        int k = idx & 1023;
        As[m * AS_STRIDE + k] = lhs[(m * T_ + (T_ - 1)) * D_ + k];
    }
    if (tid < 128) Sred[tid] = 0.0f;
    __syncthreads();

    // 512 blocks: blocks [0,256) -> h=0, [256,512) -> h=1. 8 waves x 2 tiles each.
    const int h     = blockIdx.x >> 8;
    const int vbase = (blockIdx.x & 255) * 16 + wave * 2;

    const int  m     = lane & 15;          // fragment row (A: batch, B: w-row-in-tile = r)
    const int  khalf = (lane >> 4) * 4;    // K sub-offset for upper half-wave
    const bool lo    = (lane < 16);

    // A rows 8..15 feed only C lanes 16..31, which every consumer masks out.
    // So lanes with m >= 8 just re-read row m&7 (finite garbage in dead lanes).
    const float* arow = As + (m & 7) * AS_STRIDE;

    // hoist points for E-capture
    long long pt[8];
    #pragma unroll
    for (int g = 0; g < 8; ++g) pt[g] = points[g * H_ + h];

    v8f accS = {0.f,0.f,0.f,0.f,0.f,0.f,0.f,0.f};

    for (int tt = 0; tt < 2; ++tt) {
        const int v = vbase + tt;
        const long long nbase = (long long)(h * V_ + v) * R_;   // first w-row of tile
        const float* wrow = w + (nbase + m) * D_;               // this lane's weight row

        v8f c = {0.f,0.f,0.f,0.f,0.f,0.f,0.f,0.f};
        #pragma unroll 8
        for (int j = 0; j < 128; ++j) {
            const int col = j * 8 + khalf;                      // K permuted in groups of 8
            // streaming weights: non-temporal so 512MB doesn't trash L2
            f32x4 bf = __builtin_nontemporal_load((const f32x4*)(wrow + col));
            f32x4 af = *(const f32x4*)(arow + col);             // ds_load_b128
            v2f a0 = {af.x, af.y}, b0 = {bf.x, bf.y};
            v2f a1 = {af.z, af.w}, b1 = {bf.z, bf.w};
            // D = A(16x4 f32) x B(4x16 f32) + C   (fp32 WMMA, wave32)
            c = __builtin_amdgcn_wmma_f32_16x16x4_f32(false, a0, false, b0,
                                                      (short)0, c, false, false);
            c = __builtin_amdgcn_wmma_f32_16x16x4_f32(false, a1, false, b1,
                                                      (short)0, c, false, false);
        }

        // C layout: VGPR g, lanes 0..15 -> (batch=g, r=lane). Fuse bias+exp+reduce.
        const float bv = bias[nbase + m];   // column index == lane&15 == r
        #pragma unroll
        for (int g = 0; g < 8; ++g) {
            float p = lo ? expf(c[g] + bv) : 0.0f;
            accS[g] += p;
            if (lo && (long long)v == pt[g])            // E capture for p_eval
                ws[256 + h * 128 + g * 16 + lane] = p;
        }
    }

    // two-level S reduction: registers -> LDS atomics -> global atomics
    if (lo) {
        #pragma unroll
        for (int g = 0; g < 8; ++g)
            atomicAdd(&Sred[g * 16 + lane], accS[g]);
    }
    __syncthreads();
    if (tid < 128)
        atomicAdd(&ws[h * 128 + tid], Sred[tid]);
}

__global__ void cp_finalize(const float* __restrict__ ws, float* __restrict__ out) {
    int b = threadIdx.x;
    if (b < 8) {
        const float* S0 = ws + 0 * 128 + b * 16;
        const float* S1 = ws + 1 * 128 + b * 16;
        const float* E0 = ws + 256 + 0 * 128 + b * 16;
        const float* E1 = ws + 256 + 1 * 128 + b * 16;
        float nc = 0.f, pe = 0.f;
        #pragma unroll
        for (int r = 0; r < 16; ++r) {
            nc += S0[r] * S1[r];
            pe += E0[r] * E1[r];
        }
        out[b]     = pe;   // p_eval
        out[8 + b] = nc;   // norm_const
    }
}

extern "C" void kernel_launch(void* const* d_in, const int* in_sizes, int n_in,
                              void* d_out, int out_size, void* d_ws, size_t ws_size,
                              hipStream_t stream) {
    const float*     lhs    = (const float*)d_in[0];
    const float*     w      = (const float*)d_in[1];
    const float*     bias   = (const float*)d_in[2];
    const long long* points = (const long long*)d_in[3];
    float* ws  = (float*)d_ws;
    float* out = (float*)d_out;

    cp_zero_ws<<<1, 512, 0, stream>>>(ws);
    cp_main<<<512, 256, 0, stream>>>(lhs, w, bias, points, ws);
    cp_finalize<<<1, 64, 0, stream>>>(ws, out);
}